// GlobalContrastiveLoss_24335284699669
// MI455X (gfx1250) — compile-verified
//
#include <hip/hip_runtime.h>
#include <hip/hip_bf16.h>

typedef __attribute__((ext_vector_type(2))) float v2f;
typedef __attribute__((ext_vector_type(8))) float v8f;

#define INV_T 5.0f          // 1 / 0.2
#define EPS   1e-12f
#define B_DIM 1024
#define D_DIM 128
#define Q_DIM 65536

// ---------------------------------------------------------------------------
// Kernel 1: per-row L2-normalize gf1 -> qn (ws), and pos_sim = (qn . kn)/T
// One block per row b, 128 threads (one per feature).
// ---------------------------------------------------------------------------
__global__ void norm_pos_kernel(const float* __restrict__ g1,
                                const float* __restrict__ g2,
                                float* __restrict__ qn,
                                float* __restrict__ pos_out) {
    __shared__ float s1[D_DIM];
    __shared__ float s2[D_DIM];
    const int b = blockIdx.x;
    const int t = threadIdx.x;
    const float x1 = g1[b * D_DIM + t];
    const float x2 = g2[b * D_DIM + t];
    s1[t] = x1 * x1;
    s2[t] = x2 * x2;
    __syncthreads();
    for (int o = D_DIM / 2; o > 0; o >>= 1) {
        if (t < o) { s1[t] += s1[t + o]; s2[t] += s2[t + o]; }
        __syncthreads();
    }
    const float inv1 = 1.0f / fmaxf(sqrtf(s1[0]), EPS);
    const float inv2 = 1.0f / fmaxf(sqrtf(s2[0]), EPS);
    __syncthreads();                     // everyone has read s1[0]/s2[0]
    const float q = x1 * inv1;
    const float k = x2 * inv2;
    qn[b * D_DIM + t] = q;
    s1[t] = q * k;
    __syncthreads();
    for (int o = D_DIM / 2; o > 0; o >>= 1) {
        if (t < o) s1[t] += s1[t + o];
        __syncthreads();
    }
    if (t == 0) pos_out[b] = s1[0] * INV_T;
}

// ---------------------------------------------------------------------------
// Kernel 2: invq[j] = 1 / max(||queue[j]||, eps).  One wave32 per row,
// 8 rows per 256-thread block; float4 loads (4 elems/lane * 32 lanes = 128).
// ---------------------------------------------------------------------------
__global__ void queue_invnorm_kernel(const float* __restrict__ queue,
                                     float* __restrict__ invq) {
    const int lane = threadIdx.x & 31;
    const int wave = threadIdx.x >> 5;
    const int row  = blockIdx.x * 8 + wave;
    const float4 v = ((const float4*)(queue + (size_t)row * D_DIM))[lane];
    float s = v.x * v.x + v.y * v.y + v.z * v.z + v.w * v.w;
    #pragma unroll
    for (int m = 16; m > 0; m >>= 1) s += __shfl_xor(s, m, 32);
    if (lane == 0) invq[row] = 1.0f / fmaxf(sqrtf(s), EPS);
}

// ---------------------------------------------------------------------------
// Kernel 3: neg_sim = (qn @ queue^T) * invq[j] * (1/T) via V_WMMA_F32_16X16X4.
// 256 threads = 8 waves; each wave owns one 16(M) x 16(N) tile, K=128 in 32
// steps of K=4.  Fragment layouts per ISA 7.12.2 (32-bit A 16x4 / B 4x16):
//   lanes 0-15  : {K+0, K+1} for M/N = lane
//   lanes 16-31 : {K+2, K+3} for M/N = lane-16
// ---------------------------------------------------------------------------
__global__ void gemm_wmma_kernel(const float* __restrict__ qn,
                                 const float* __restrict__ queue,
                                 const float* __restrict__ invq,
                                 float* __restrict__ neg) {
    const int lane   = threadIdx.x & 31;
    const int wave   = threadIdx.x >> 5;
    const int l      = lane & 15;
    const int half   = lane >> 4;                       // 0 or 1
    const int tile_n = (blockIdx.x * 8 + wave) * 16;    // N base
    const int tile_m = blockIdx.y * 16;                 // M base

    const float* arow = qn    + (size_t)(tile_m + l) * D_DIM + 2 * half;
    const float* brow = queue + (size_t)(tile_n + l) * D_DIM + 2 * half;

    v8f acc = {};
    #pragma unroll 8
    for (int s = 0; s < 32; ++s) {
        v2f a = *(const v2f*)(arow + 4 * s);
        v2f b = *(const v2f*)(brow + 4 * s);
        acc = __builtin_amdgcn_wmma_f32_16x16x4_f32(
            /*neg_a=*/false, a, /*neg_b=*/false, b,
            /*c_mod=*/(short)0, acc, /*reuse_a=*/false, /*reuse_b=*/false);
    }

    // D layout: VGPR r -> (M = r + 8*half, N = l); scale by invq[col]/T.
    const float scale = invq[tile_n + l] * INV_T;
    float* dst = neg + (size_t)(tile_m + half * 8) * Q_DIM + tile_n + l;
    #pragma unroll
    for (int r = 0; r < 8; ++r) {
        dst[(size_t)r * Q_DIM] = acc[r] * scale;
    }
}

// ---------------------------------------------------------------------------
// Kernel 4: per-row logsumexp term, deterministic fixed-order reduction.
// One block per row; float4 streaming reads of neg_sim.
// partial[b] = log( sum_j exp(neg[b,j]) + exp(pos[b]) ) - pos[b]
// ---------------------------------------------------------------------------
__global__ void lse_kernel(const float* __restrict__ neg,
                           const float* __restrict__ pos,
                           float* __restrict__ partial) {
    __shared__ float sm[256];
    const int b = blockIdx.x;
    const int t = threadIdx.x;
    const float4* row = (const float4*)(neg + (size_t)b * Q_DIM);
    float s = 0.0f;
    for (int j = t; j < Q_DIM / 4; j += 256) {
        const float4 v = row[j];
        s += __expf(v.x) + __expf(v.y) + __expf(v.z) + __expf(v.w);
    }
    sm[t] = s;
    __syncthreads();
    for (int o = 128; o > 0; o >>= 1) {
        if (t < o) sm[t] += sm[t + o];
        __syncthreads();
    }
    if (t == 0) {
        const float p = pos[b];
        partial[b] = logf(sm[0] + __expf(p)) - p;
    }
}

// ---------------------------------------------------------------------------
// Kernel 5: loss = mean(partial)
// ---------------------------------------------------------------------------
__global__ void loss_kernel(const float* __restrict__ partial,
                            float* __restrict__ out) {
    __shared__ float sm[256];
    const int t = threadIdx.x;
    float s = partial[t] + partial[t + 256] + partial[t + 512] + partial[t + 768];
    sm[t] = s;
    __syncthreads();
    for (int o = 128; o > 0; o >>= 1) {
        if (t < o) sm[t] += sm[t + o];
        __syncthreads();
    }
    if (t == 0) out[0] = sm[0] * (1.0f / (float)B_DIM);
}

// ---------------------------------------------------------------------------
extern "C" void kernel_launch(void* const* d_in, const int* in_sizes, int n_in,
                              void* d_out, int out_size, void* d_ws, size_t ws_size,
                              hipStream_t stream) {
    const float* g1    = (const float*)d_in[0];   // [1024,128]
    const float* g2    = (const float*)d_in[1];   // [1024,128]
    const float* queue = (const float*)d_in[2];   // [65536,128]

    float* out  = (float*)d_out;
    float* loss = out;                 // [0]
    float* pos  = out + 1;             // [1 .. 1025)
    float* neg  = out + 1 + B_DIM;     // [1025 .. 1025 + 1024*65536)

    float* ws      = (float*)d_ws;
    float* qn      = ws;                               // 1024*128
    float* invq    = qn + B_DIM * D_DIM;               // 65536
    float* partial = invq + Q_DIM;                     // 1024

    // 1) normalize gf1/gf2 rows, write qn + pos_sim
    norm_pos_kernel<<<B_DIM, D_DIM, 0, stream>>>(g1, g2, qn, pos);

    // 2) inverse norms of queue rows
    queue_invnorm_kernel<<<Q_DIM / 8, 256, 0, stream>>>(queue, invq);

    // 3) WMMA GEMM with fused normalize + 1/T scale
    dim3 grid(Q_DIM / (8 * 16), B_DIM / 16, 1);        // (512, 64)
    gemm_wmma_kernel<<<grid, 256, 0, stream>>>(qn, queue, invq, neg);

    // 4) per-row logsumexp - pos  (deterministic)
    lse_kernel<<<B_DIM, 256, 0, stream>>>(neg, pos, partial);

    // 5) mean -> loss
    loss_kernel<<<1, 256, 0, stream>>>(partial, loss);
}